// GraphConv_54803782697379
// MI455X (gfx1250) — compile-verified
//
#include <hip/hip_runtime.h>

#define N_AGENTS 64
#define N_EDGES  4032
#define D_NODE   16
#define HDIM     64
#define BATCH    128

typedef __attribute__((ext_vector_type(2))) float v2f;
typedef __attribute__((ext_vector_type(8))) float v8f;

// D = A(16x4 f32) * B(4x16 f32) + C(16x16 f32), wave32 matrix op
__device__ __forceinline__ v8f wmma4(v2f a, v2f b, v8f c) {
    return __builtin_amdgcn_wmma_f32_16x16x4_f32(
        /*neg_a=*/false, a, /*neg_b=*/false, b,
        /*c_mod=*/(short)0, c, /*reuse_a=*/false, /*reuse_b=*/false);
}

// ---------------------------------------------------------------------------
// Kernel 1: per (batch, receiver) workgroup.
// 64 "edge rows" (senders 0..63, self-loop masked to zero), two 2-layer MLPs
// via f32 WMMA, masked by edge-type weights, reduced over rows into the
// 64-wide node message. 4 waves x 16-row tiles.
// ---------------------------------------------------------------------------
__global__ __launch_bounds__(128)
void edge_encode_scatter(const float* __restrict__ ns,
                         const float* __restrict__ et,
                         const float* __restrict__ e0w1, const float* __restrict__ e0b1,
                         const float* __restrict__ e0w2, const float* __restrict__ e0b2,
                         const float* __restrict__ e1w1, const float* __restrict__ e1b1,
                         const float* __restrict__ e1w2, const float* __restrict__ e1b2,
                         float* __restrict__ node_msg)
{
    __shared__ float s_w1[32 * 64];      // current encoder layer-1 weights
    __shared__ float s_w2[64 * 64];      // current encoder layer-2 weights
    __shared__ float s_b[4 * 64];        // e0b1 | e0b2 | e1b1 | e1b2
    __shared__ float s_t1[64];           // edge-type-1 weight per sender row
    __shared__ float s_t2[64];           // edge-type-2 weight per sender row
    __shared__ float s_x[64][33];        // edge inputs [row][32], padded
    __shared__ float s_h[64][68];        // layer-1 activations, per-wave 16-row slabs
    __shared__ float s_msg[64];          // reduced node message

    const int b    = blockIdx.x >> 6;
    const int r    = blockIdx.x & 63;
    const int tid  = threadIdx.x;
    const int wave = tid >> 5;
    const int lane = tid & 31;
    const int lh   = lane >> 4;          // 0: K=0,1 half / rows 0-7 of C ; 1: K=2,3 / rows 8-15
    const int ln   = lane & 15;
    const int rowA = (wave << 4) + ln;   // matrix row this lane's A-fragment covers

    // ---- stage encoder-0 weights + shared data ----
    for (int i = tid; i < 32 * 64; i += 128) s_w1[i] = e0w1[i];
    for (int i = tid; i < 64 * 64; i += 128) s_w2[i] = e0w2[i];
    if (tid < 64) {
        s_b[tid]        = e0b1[tid];
        s_b[64  + tid]  = e0b2[tid];
        s_b[128 + tid]  = e1b1[tid];
        s_b[192 + tid]  = e1b2[tid];
        // edge-type scalars for sender s = tid -> receiver r (self-loop: zero)
        float t1 = 0.f, t2 = 0.f;
        const int s = tid;
        if (s != r) {
            const int e = s * 63 + (r < s ? r : r - 1);     // np.where row-major edge id
            const float* p = et + ((size_t)b * N_EDGES + e) * 3;
            t1 = p[1];
            t2 = p[2];
        }
        s_t1[s]  = t1;
        s_t2[s]  = t2;
        s_msg[s] = 0.f;
    }
    {   // x[row] = [ ns[b, sender=row, :] , ns[b, r, :] ]
        const int row  = tid >> 1;
        const int half = tid & 1;
        const float* src = ns + ((size_t)b * N_AGENTS + (half ? r : row)) * D_NODE;
        #pragma unroll
        for (int c = 0; c < 16; ++c) s_x[row][half * 16 + c] = src[c];
    }
    __syncthreads();

    v8f y0[4], y1[4];

    // ================= encoder 0 =================
    #pragma unroll
    for (int nt = 0; nt < 4; ++nt) {                 // layer 1: [16x32]@[32x64]
        v8f c = {};
        #pragma unroll
        for (int k4 = 0; k4 < 8; ++k4) {
            const int kb = k4 * 4 + 2 * lh;
            v2f a = { s_x[rowA][kb], s_x[rowA][kb + 1] };
            v2f w = { s_w1[kb * 64 + nt * 16 + ln], s_w1[(kb + 1) * 64 + nt * 16 + ln] };
            c = wmma4(a, w, c);
        }
        const float bias = s_b[nt * 16 + ln];
        #pragma unroll
        for (int v = 0; v < 8; ++v) {
            float t = c[v] + bias;
            s_h[(wave << 4) + v + 8 * lh][nt * 16 + ln] = t > 0.f ? t : 0.f;
        }
    }
    #pragma unroll
    for (int nt = 0; nt < 4; ++nt) {                 // layer 2: [16x64]@[64x64]
        v8f c = {};
        #pragma unroll
        for (int k4 = 0; k4 < 16; ++k4) {
            const int kb = k4 * 4 + 2 * lh;
            v2f a = { s_h[rowA][kb], s_h[rowA][kb + 1] };
            v2f w = { s_w2[kb * 64 + nt * 16 + ln], s_w2[(kb + 1) * 64 + nt * 16 + ln] };
            c = wmma4(a, w, c);
        }
        const float bias = s_b[64 + nt * 16 + ln];
        #pragma unroll
        for (int v = 0; v < 8; ++v) {
            float t = c[v] + bias;
            y0[nt][v] = t > 0.f ? t : 0.f;
        }
    }
    __syncthreads();

    // ---- swap in encoder-1 weights ----
    for (int i = tid; i < 32 * 64; i += 128) s_w1[i] = e1w1[i];
    for (int i = tid; i < 64 * 64; i += 128) s_w2[i] = e1w2[i];
    __syncthreads();

    // ================= encoder 1 =================
    #pragma unroll
    for (int nt = 0; nt < 4; ++nt) {
        v8f c = {};
        #pragma unroll
        for (int k4 = 0; k4 < 8; ++k4) {
            const int kb = k4 * 4 + 2 * lh;
            v2f a = { s_x[rowA][kb], s_x[rowA][kb + 1] };
            v2f w = { s_w1[kb * 64 + nt * 16 + ln], s_w1[(kb + 1) * 64 + nt * 16 + ln] };
            c = wmma4(a, w, c);
        }
        const float bias = s_b[128 + nt * 16 + ln];
        #pragma unroll
        for (int v = 0; v < 8; ++v) {
            float t = c[v] + bias;
            s_h[(wave << 4) + v + 8 * lh][nt * 16 + ln] = t > 0.f ? t : 0.f;
        }
    }
    #pragma unroll
    for (int nt = 0; nt < 4; ++nt) {
        v8f c = {};
        #pragma unroll
        for (int k4 = 0; k4 < 16; ++k4) {
            const int kb = k4 * 4 + 2 * lh;
            v2f a = { s_h[rowA][kb], s_h[rowA][kb + 1] };
            v2f w = { s_w2[kb * 64 + nt * 16 + ln], s_w2[(kb + 1) * 64 + nt * 16 + ln] };
            c = wmma4(a, w, c);
        }
        const float bias = s_b[192 + nt * 16 + ln];
        #pragma unroll
        for (int v = 0; v < 8; ++v) {
            float t = c[v] + bias;
            y1[nt][v] = t > 0.f ? t : 0.f;
        }
    }

    // ---- masked row-reduction: msg[N] = sum_rows( y0*t1 + y1*t2 ) ----
    #pragma unroll
    for (int nt = 0; nt < 4; ++nt) {
        float p = 0.f;
        #pragma unroll
        for (int v = 0; v < 8; ++v) {
            const int j = (wave << 4) + v + 8 * lh;      // edge row (== sender id)
            p += y0[nt][v] * s_t1[j] + y1[nt][v] * s_t2[j];
        }
        atomicAdd(&s_msg[nt * 16 + ln], p);              // ds_add_f32
    }
    __syncthreads();

    if (tid < 64) node_msg[(size_t)blockIdx.x * 64 + tid] = s_msg[tid];
}

// ---------------------------------------------------------------------------
// Kernel 2: node decoder. 8192 rows -> MLP(80->64->16) via WMMA.
// 128 blocks x 64 rows; one 16-row tile per wave.
// ---------------------------------------------------------------------------
__global__ __launch_bounds__(128)
void node_decode(const float* __restrict__ ns,
                 const float* __restrict__ msg,
                 const float* __restrict__ w1, const float* __restrict__ b1,
                 const float* __restrict__ w2, const float* __restrict__ b2,
                 float* __restrict__ out)
{
    __shared__ float s_w1[80 * 64];
    __shared__ float s_w2[64 * 16];
    __shared__ float s_b1[64];
    __shared__ float s_b2[16];
    __shared__ float s_z[64][81];        // [row][80] inputs, padded
    __shared__ float s_h[64][68];        // layer-1 activations

    const int tid  = threadIdx.x;
    const int wave = tid >> 5;
    const int lane = tid & 31;
    const int lh   = lane >> 4;
    const int ln   = lane & 15;
    const int rowA = (wave << 4) + ln;
    const int base = blockIdx.x * 64;    // global row base (row = b*64 + n)

    for (int i = tid; i < 80 * 64; i += 128) s_w1[i] = w1[i];
    for (int i = tid; i < 64 * 16; i += 128) s_w2[i] = w2[i];
    if (tid < 64) s_b1[tid] = b1[tid];
    if (tid < 16) s_b2[tid] = b2[tid];
    for (int i = tid; i < 64 * 80; i += 128) {
        const int lr = i / 80, c = i % 80;
        const int row = base + lr;
        s_z[lr][c] = (c < 16) ? ns[(size_t)row * 16 + c]
                              : msg[(size_t)row * 64 + (c - 16)];
    }
    __syncthreads();

    // layer 1: [16x80]@[80x64]
    #pragma unroll
    for (int nt = 0; nt < 4; ++nt) {
        v8f c = {};
        #pragma unroll
        for (int k4 = 0; k4 < 20; ++k4) {
            const int kb = k4 * 4 + 2 * lh;
            v2f a = { s_z[rowA][kb], s_z[rowA][kb + 1] };
            v2f w = { s_w1[kb * 64 + nt * 16 + ln], s_w1[(kb + 1) * 64 + nt * 16 + ln] };
            c = wmma4(a, w, c);
        }
        const float bias = s_b1[nt * 16 + ln];
        #pragma unroll
        for (int v = 0; v < 8; ++v) {
            float t = c[v] + bias;
            s_h[(wave << 4) + v + 8 * lh][nt * 16 + ln] = t > 0.f ? t : 0.f;
        }
    }
    // layer 2: [16x64]@[64x16]
    {
        v8f c = {};
        #pragma unroll
        for (int k4 = 0; k4 < 16; ++k4) {
            const int kb = k4 * 4 + 2 * lh;
            v2f a = { s_h[rowA][kb], s_h[rowA][kb + 1] };
            v2f w = { s_w2[kb * 16 + ln], s_w2[(kb + 1) * 16 + ln] };
            c = wmma4(a, w, c);
        }
        const float bias = s_b2[ln];
        #pragma unroll
        for (int v = 0; v < 8; ++v) {
            float t = c[v] + bias;
            const int row = base + (wave << 4) + v + 8 * lh;
            out[(size_t)row * 16 + ln] = t > 0.f ? t : 0.f;
        }
    }
}

extern "C" void kernel_launch(void* const* d_in, const int* in_sizes, int n_in,
                              void* d_out, int out_size, void* d_ws, size_t ws_size,
                              hipStream_t stream) {
    const float* ns   = (const float*)d_in[0];
    const float* et   = (const float*)d_in[1];
    const float* e0w1 = (const float*)d_in[2];
    const float* e0b1 = (const float*)d_in[3];
    const float* e0w2 = (const float*)d_in[4];
    const float* e0b2 = (const float*)d_in[5];
    const float* e1w1 = (const float*)d_in[6];
    const float* e1b1 = (const float*)d_in[7];
    const float* e1w2 = (const float*)d_in[8];
    const float* e1b2 = (const float*)d_in[9];
    const float* ndw1 = (const float*)d_in[10];
    const float* ndb1 = (const float*)d_in[11];
    const float* ndw2 = (const float*)d_in[12];
    const float* ndb2 = (const float*)d_in[13];

    float* msg = (float*)d_ws;  // [BATCH*N_AGENTS][64] = 2 MB scratch

    edge_encode_scatter<<<BATCH * N_AGENTS, 128, 0, stream>>>(
        ns, et, e0w1, e0b1, e0w2, e0b2, e1w1, e1b1, e1w2, e1b2, msg);

    node_decode<<<128, 128, 0, stream>>>(
        ns, msg, ndw1, ndb1, ndw2, ndb2, (float*)d_out);
}